// ChamferLoss_9268539424789
// MI455X (gfx1250) — compile-verified
//
#include <hip/hip_runtime.h>
#include <hip/hip_bf16.h>
#include <math.h>

// Chamfer loss for B=4, N=M=4096, D=3 (matches reference setup_inputs()).
// Cross term computed with V_WMMA_F32_16X16X4_F32 (K=4, D padded 3->4).
// min_n sqrt(max(0, a2+b2-2c)) == sqrt(max(0, a2 + min_n (b2-2c)))  (monotone),
// so sqrt/clamp/a2-add are all hoisted out of the sweep loop.

typedef __attribute__((ext_vector_type(2))) float v2f;
typedef __attribute__((ext_vector_type(8))) float v8f;

#define TILE 16
#define TILES_PER_WAVE 2
#define ROWS_PER_WAVE (TILE * TILES_PER_WAVE)          // 32
#define WAVES_PER_BLOCK 8
#define BLOCK_THREADS (WAVES_PER_BLOCK * 32)           // 256
#define ROWS_PER_BLOCK (WAVES_PER_BLOCK * ROWS_PER_WAVE) // 256
#define MAX_LB 4096

// For each row a of A (length LA, D=3), compute min over all rows b of B
// (length LB) of ||A[a]-B[b]||, per batch. outmin[batch*LA + a].
__global__ __launch_bounds__(BLOCK_THREADS)
void chamfer_min_kernel(const float* __restrict__ A,
                        const float* __restrict__ Bm,
                        float* __restrict__ outmin,
                        int LA, int LB)
{
    __shared__ float sB[MAX_LB * 3];   // 48 KB: whole B operand for this batch

    const int batch = blockIdx.y;
    const float* Abase = A  + (size_t)batch * LA * 3;
    const float* Bbase = Bm + (size_t)batch * LB * 3;

    const int tid = threadIdx.x;
    for (int i = tid; i < LB * 3; i += BLOCK_THREADS)
        sB[i] = Bbase[i];
    __syncthreads();

    const int wave    = tid >> 5;
    const int lane    = tid & 31;
    const int l15     = lane & 15;
    const int halfsel = lane >> 4;            // 0: lanes 0-15, 1: lanes 16-31
    const int m0      = (blockIdx.x * WAVES_PER_BLOCK + wave) * ROWS_PER_WAVE;

    // ---- two A fragments (each 16x4 f32, K=3 padded with 0) ----
    // lane<16 : VGPR0=K0, VGPR1=K1 of row M=l15
    // lane>=16: VGPR0=K2, VGPR1=K3(=0) of row M=l15
    const int arow0 = m0 + l15;
    const int arow1 = m0 + TILE + l15;
    const float ax0 = Abase[arow0 * 3 + 0];
    const float ay0 = Abase[arow0 * 3 + 1];
    const float az0 = Abase[arow0 * 3 + 2];
    const float ax1 = Abase[arow1 * 3 + 0];
    const float ay1 = Abase[arow1 * 3 + 1];
    const float az1 = Abase[arow1 * 3 + 2];
    const float a2_mine0 = ax0 * ax0 + ay0 * ay0 + az0 * az0;
    const float a2_mine1 = ax1 * ax1 + ay1 * ay1 + az1 * az1;

    v2f afrag0, afrag1;
    afrag0.x = halfsel ? az0 : ax0;
    afrag0.y = halfsel ? 0.0f : ay0;
    afrag1.x = halfsel ? az1 : ax1;
    afrag1.y = halfsel ? 0.0f : ay1;

    // C/D layout: this lane holds rows (v + 8*halfsel), v=0..7, column l15.
    // Row r's |a|^2 lives in lane r (r in 0..15).
    float a2row0[8], a2row1[8];
#pragma unroll
    for (int v = 0; v < 8; ++v) {
        a2row0[v] = __shfl(a2_mine0, v + 8 * halfsel, 32);
        a2row1[v] = __shfl(a2_mine1, v + 8 * halfsel, 32);
    }

    v8f min0, min1;
#pragma unroll
    for (int v = 0; v < 8; ++v) { min0[v] = 3.4e38f; min1[v] = 3.4e38f; }

    // ---- sweep all B rows in 16-wide tiles: 2 WMMAs per shared B fragment ----
#pragma unroll 4
    for (int n0 = 0; n0 < LB; n0 += TILE) {
        const int nrow = n0 + l15;
        const float bx = sB[nrow * 3 + 0];
        const float by = sB[nrow * 3 + 1];
        const float bz = sB[nrow * 3 + 2];
        const float b2 = bx * bx + by * by + bz * bz;   // column n0+l15

        v2f bfrag;                                      // B (4x16), N=l15
        bfrag.x = halfsel ? bz : bx;
        bfrag.y = halfsel ? 0.0f : by;

        v8f c0 = {0.f, 0.f, 0.f, 0.f, 0.f, 0.f, 0.f, 0.f};
        v8f c1 = {0.f, 0.f, 0.f, 0.f, 0.f, 0.f, 0.f, 0.f};
        c0 = __builtin_amdgcn_wmma_f32_16x16x4_f32(
                 false, afrag0, false, bfrag, (short)0, c0, false, false);
        c1 = __builtin_amdgcn_wmma_f32_16x16x4_f32(
                 false, afrag1, false, bfrag, (short)0, c1, false, false);

#pragma unroll
        for (int v = 0; v < 8; ++v) {
            // t = b2 - 2*cross ; running min (a2 added after the loop)
            min0[v] = fminf(min0[v], __builtin_fmaf(-2.0f, c0[v], b2));
            min1[v] = fminf(min1[v], __builtin_fmaf(-2.0f, c1[v], b2));
        }
    }

    // ---- min across the 16 lanes holding the same rows ----
#pragma unroll
    for (int off = 1; off < 16; off <<= 1) {
#pragma unroll
        for (int v = 0; v < 8; ++v) {
            min0[v] = fminf(min0[v], __shfl_xor(min0[v], off, 32));
            min1[v] = fminf(min1[v], __shfl_xor(min1[v], off, 32));
        }
    }

    if (l15 == 0) {
        float* dst0 = outmin + (size_t)batch * LA + m0 + 8 * halfsel;
        float* dst1 = dst0 + TILE;
#pragma unroll
        for (int v = 0; v < 8; ++v) {
            const float d20 = fmaxf(a2row0[v] + min0[v], 0.0f);
            const float d21 = fmaxf(a2row1[v] + min1[v], 0.0f);
            dst0[v] = __builtin_sqrtf(d20);
            dst1[v] = __builtin_sqrtf(d21);
        }
    }
}

// out[0] = mean(dist1) + mean(dist2)
__global__ __launch_bounds__(256)
void chamfer_reduce_kernel(const float* __restrict__ dist1,
                           const float* __restrict__ dist2,
                           float* __restrict__ out, int n1, int n2)
{
    __shared__ float red[256];
    const int tid = threadIdx.x;
    float s1 = 0.f, s2 = 0.f;
    for (int i = tid; i < n1; i += 256) s1 += dist1[i];
    for (int i = tid; i < n2; i += 256) s2 += dist2[i];
    red[tid] = s1 / (float)n1 + s2 / (float)n2;
    __syncthreads();
    for (int s = 128; s > 0; s >>= 1) {
        if (tid < s) red[tid] += red[tid + s];
        __syncthreads();
    }
    if (tid == 0) out[0] = red[0];
}

extern "C" void kernel_launch(void* const* d_in, const int* in_sizes, int n_in,
                              void* d_out, int out_size, void* d_ws, size_t ws_size,
                              hipStream_t stream)
{
    const float* x = (const float*)d_in[0];   // [B, N, 3]
    const float* y = (const float*)d_in[1];   // [B, M, 3]

    const int B = 4;
    const int D = 3;
    const int N = in_sizes[0] / (B * D);      // 4096
    const int M = in_sizes[1] / (B * D);      // 4096

    float* dist1 = (float*)d_ws;              // [B, M]  min over n for each y row
    float* dist2 = dist1 + (size_t)B * M;     // [B, N]  min over m for each x row
    float* out   = (float*)d_out;

    dim3 block(BLOCK_THREADS);

    // dist1[b,m] = min_n ||y[b,m]-x[b,n]||  : A=y, B-operand=x
    dim3 grid1(M / ROWS_PER_BLOCK, B);
    chamfer_min_kernel<<<grid1, block, 0, stream>>>(y, x, dist1, M, N);

    // dist2[b,n] = min_m ||x[b,n]-y[b,m]||  : A=x, B-operand=y
    dim3 grid2(N / ROWS_PER_BLOCK, B);
    chamfer_min_kernel<<<grid2, block, 0, stream>>>(x, y, dist2, N, M);

    chamfer_reduce_kernel<<<1, 256, 0, stream>>>(dist1, dist2, out, B * M, B * N);
}